// BboxTargetLayer_78632261255867
// MI455X (gfx1250) — compile-verified
//
#include <hip/hip_runtime.h>
#include <hip/hip_bf16.h>
#include <math.h>

// ---------------------------------------------------------------------------
// Problem constants (match reference)
// ---------------------------------------------------------------------------
#define BB     32          // batch
#define GG     50          // num GT boxes
#define TOTAL_ANCH 36864   // FEAT*FEAT*A = 64*64*9
#define NEG_OV 0.3f
#define POS_OV 0.7f
#define NUM_FG 128         // int(0.5*256)
#define RPN_BS 256
#define NBIN   1024        // histogram bins for top-K selection

typedef unsigned int u32x4 __attribute__((ext_vector_type(4)));
typedef int          i32x8 __attribute__((ext_vector_type(8)));
typedef int          i32x4 __attribute__((ext_vector_type(4)));

// ---------------------------------------------------------------------------
// Tensor Data Mover: stage the per-batch GT tile (50x5 f32 = 250 dwords) into
// LDS.  D# built per CDNA5 ISA ch.8: group0 = {count|flags, lds_addr,
// global_addr[31:0], global_addr[56:32]|type=2}; group1 carries data_size=4B,
// tensor_dim0=250, tensor_dim1=1, tile_dim0=250, tile_dim1=1, dim0_stride=250.
// Issued by wave 0 only (TDM ignores EXEC); completion via s_wait_tensorcnt.
// This toolchain's builtin is the 6-arg form:
//   (u32x4 g0, i32x8 g1, i32x4 g2, i32x4 g3, i32x8 g4, i32 cpol)
// ---------------------------------------------------------------------------
__device__ __forceinline__ void tdm_load_gt(const float* __restrict__ src,
                                            float* __restrict__ sgt) {
#if __has_builtin(__builtin_amdgcn_tensor_load_to_lds)
  if (threadIdx.x < 32) {
    unsigned long long ga = (unsigned long long)(const void*)src;
    unsigned lds = (unsigned)(unsigned long long)(void*)sgt;  // low 32 bits = LDS offset
    u32x4 g0;
    g0.x = 1u;                                         // count=1 (valid user D#)
    g0.y = lds;                                        // lds_addr (bytes)
    g0.z = (unsigned)(ga & 0xFFFFFFFFull);             // global_addr[31:0]
    g0.w = ((unsigned)((ga >> 32) & 0x01FFFFFFull))    // global_addr[56:32]
           | 0x80000000u;                              // type=2 ("image") at [127:126]
    i32x8 g1;
    g1[0] = 0x00020000;                 // workgroup_mask=0, data_size=2 (4 bytes)
    g1[1] = (int)(250u << 16);          // tensor_dim0[15:0]=250 at bits [63:48]
    g1[2] = (int)(1u << 16);            // tensor_dim0 hi=0 | tensor_dim1=1
    g1[3] = (int)(250u << 16);          // tensor_dim1 hi=0 | tile_dim0=250
    g1[4] = 1;                          // tile_dim1=1, tile_dim2=0
    g1[5] = 250;                        // tensor_dim0_stride low32 = 250
    g1[6] = 0;                          // stride hi, dim1_stride lo (unused, 2D)
    g1[7] = 0;
    i32x4 z4 = {0, 0, 0, 0};            // groups 2/3 unused (<=2D)
    i32x8 z8 = {0, 0, 0, 0, 0, 0, 0, 0};
    __builtin_amdgcn_tensor_load_to_lds(g0, g1, z4, z4, z8, 0);
    __builtin_amdgcn_s_wait_tensorcnt(0);
  }
#else
  for (int t = threadIdx.x; t < GG * 5; t += blockDim.x) sgt[t] = src[t];
#endif
  __syncthreads();
}

// ---------------------------------------------------------------------------
// Shared IoU (identical code in max pass and equality pass)
// ---------------------------------------------------------------------------
__device__ __forceinline__ float iou_ov(float a0, float a1, float a2, float a3,
                                        float aarea,
                                        float g0, float g1, float g2, float g3) {
  float gw = g2 - g0 + 1.0f, gh = g3 - g1 + 1.0f;
  float garea = gw * gh;
  float iw = fminf(a2, g2) - fmaxf(a0, g0) + 1.0f;
  float ih = fminf(a3, g3) - fmaxf(a1, g1) + 1.0f;
  float inter = fmaxf(iw, 0.0f) * fmaxf(ih, 0.0f);
  float ov = inter / (aarea + garea - inter);
  return (garea == 0.0f) ? 0.0f : ov;
}

// ---------------------------------------------------------------------------
// JAX Threefry-2x32 (rotations/key-schedule match jax._src.prng)
// ---------------------------------------------------------------------------
__device__ __forceinline__ unsigned rotl32(unsigned x, int d) {
  return (x << d) | (x >> (32 - d));
}
__device__ __forceinline__ void tf2x32(unsigned k0, unsigned k1,
                                       unsigned& x0, unsigned& x1) {
  const unsigned ks2 = k0 ^ k1 ^ 0x1BD11BDAu;
  x0 += k0; x1 += k1;
  x0 += x1; x1 = rotl32(x1, 13); x1 ^= x0;
  x0 += x1; x1 = rotl32(x1, 15); x1 ^= x0;
  x0 += x1; x1 = rotl32(x1, 26); x1 ^= x0;
  x0 += x1; x1 = rotl32(x1, 6);  x1 ^= x0;
  x0 += k1; x1 += ks2 + 1u;
  x0 += x1; x1 = rotl32(x1, 17); x1 ^= x0;
  x0 += x1; x1 = rotl32(x1, 29); x1 ^= x0;
  x0 += x1; x1 = rotl32(x1, 16); x1 ^= x0;
  x0 += x1; x1 = rotl32(x1, 24); x1 ^= x0;
  x0 += ks2; x1 += k0 + 2u;
  x0 += x1; x1 = rotl32(x1, 13); x1 ^= x0;
  x0 += x1; x1 = rotl32(x1, 15); x1 ^= x0;
  x0 += x1; x1 = rotl32(x1, 26); x1 ^= x0;
  x0 += x1; x1 = rotl32(x1, 6);  x1 ^= x0;
  x0 += k0; x1 += k1 + 3u;
  x0 += x1; x1 = rotl32(x1, 17); x1 ^= x0;
  x0 += x1; x1 = rotl32(x1, 29); x1 ^= x0;
  x0 += x1; x1 = rotl32(x1, 16); x1 ^= x0;
  x0 += x1; x1 = rotl32(x1, 24); x1 ^= x0;
  x0 += k1; x1 += ks2 + 4u;
  x0 += x1; x1 = rotl32(x1, 13); x1 ^= x0;
  x0 += x1; x1 = rotl32(x1, 15); x1 ^= x0;
  x0 += x1; x1 = rotl32(x1, 26); x1 ^= x0;
  x0 += x1; x1 = rotl32(x1, 6);  x1 ^= x0;
  x0 += ks2; x1 += k0 + 5u;
}

// bits for element i of random_bits(key,(N,)) with JAX's split-halves pairing
__device__ __forceinline__ float tf_uniform(unsigned k0, unsigned k1, int i, int N) {
  unsigned h = ((unsigned)N + 1u) >> 1;  // ceil(N/2) (odd N zero-padded)
  unsigned c0, c1; bool hi;
  if ((unsigned)i < h) {
    c0 = (unsigned)i;
    unsigned p = h + (unsigned)i;
    c1 = (p < (unsigned)N) ? p : 0u;
    hi = false;
  } else {
    c0 = (unsigned)i - h; c1 = (unsigned)i; hi = true;
  }
  unsigned x0 = c0, x1 = c1;
  tf2x32(k0, k1, x0, x1);
  unsigned bits = hi ? x1 : x0;
  float f = __uint_as_float((bits >> 9) | 0x3F800000u) - 1.0f;  // [0,1)
  return fmaxf(f, 0.0f);
}

// ---------------------------------------------------------------------------
// K0: init outputs (labels=-1, targets=0), zero gt_max + histograms
// ---------------------------------------------------------------------------
__global__ void k_init(float* __restrict__ out, int out_size,
                       unsigned* __restrict__ gtmax, int* __restrict__ hist) {
  for (int idx = blockIdx.x * blockDim.x + threadIdx.x; idx < out_size;
       idx += gridDim.x * blockDim.x) {
    out[idx] = (idx < BB * TOTAL_ANCH) ? -1.0f : 0.0f;
    if (idx < BB * GG) gtmax[idx] = 0u;
    if (idx < BB * 2 * NBIN) hist[idx] = 0;
  }
}

// ---------------------------------------------------------------------------
// K1: per-anchor max/argmax over GT + per-GT max over anchors
// ---------------------------------------------------------------------------
__global__ void k_overlap(const float* __restrict__ gt,
                          const float* __restrict__ anchors,
                          float* __restrict__ maxov, int* __restrict__ amax,
                          unsigned* __restrict__ gtmax, int N) {
  __shared__ float sgt[GG * 5];
  __shared__ unsigned sgmax[GG];
  const int b = blockIdx.y;
  tdm_load_gt(gt + (size_t)b * GG * 5, sgt);
  if (threadIdx.x < GG) sgmax[threadIdx.x] = 0u;
  __syncthreads();

  const int i = blockIdx.x * blockDim.x + threadIdx.x;
  if (i < N) {
    float a0 = anchors[4 * i + 0], a1 = anchors[4 * i + 1];
    float a2 = anchors[4 * i + 2], a3 = anchors[4 * i + 3];
    float aw = a2 - a0 + 1.0f, ah = a3 - a1 + 1.0f;
    float aarea = aw * ah;
    float best = -1e30f; int bi = 0;
    for (int g = 0; g < GG; ++g) {
      float ov = iou_ov(a0, a1, a2, a3, aarea,
                        sgt[5 * g], sgt[5 * g + 1], sgt[5 * g + 2], sgt[5 * g + 3]);
      if (ov > best) { best = ov; bi = g; }           // argmax keeps first max
      atomicMax(&sgmax[g], __float_as_uint(ov));      // ov>=0: bit order == value order
    }
    maxov[(size_t)b * N + i] = best;
    amax[(size_t)b * N + i]  = bi;
  }
  __syncthreads();
  if (threadIdx.x < GG)
    atomicMax(&gtmax[b * GG + threadIdx.x], sgmax[threadIdx.x]);
}

// ---------------------------------------------------------------------------
// K2: label assignment (-1 / 0 / 1) incl. gt-argmax "keep" anchors
// ---------------------------------------------------------------------------
__global__ void k_labels(const float* __restrict__ gt,
                         const float* __restrict__ anchors,
                         const float* __restrict__ maxov,
                         const unsigned* __restrict__ gtmax,
                         float* __restrict__ labels, int N) {
  __shared__ float sgt[GG * 5];
  __shared__ float sgadj[GG];
  const int b = blockIdx.y;
  tdm_load_gt(gt + (size_t)b * GG * 5, sgt);
  if (threadIdx.x < GG) {
    float m = __uint_as_float(gtmax[b * GG + threadIdx.x]);
    sgadj[threadIdx.x] = (m == 0.0f) ? 1e-5f : m;
  }
  __syncthreads();

  const int i = blockIdx.x * blockDim.x + threadIdx.x;
  if (i >= N) return;
  float a0 = anchors[4 * i + 0], a1 = anchors[4 * i + 1];
  float a2 = anchors[4 * i + 2], a3 = anchors[4 * i + 3];
  float aw = a2 - a0 + 1.0f, ah = a3 - a1 + 1.0f;
  float aarea = aw * ah;

  float lab = -1.0f;
  float mo = maxov[(size_t)b * N + i];
  if (mo < NEG_OV) lab = 0.0f;
  int keep = 0;
  for (int g = 0; g < GG; ++g) {
    float ov = iou_ov(a0, a1, a2, a3, aarea,
                      sgt[5 * g], sgt[5 * g + 1], sgt[5 * g + 2], sgt[5 * g + 3]);
    keep += (ov == sgadj[g]);
  }
  if (keep > 0) lab = 1.0f;
  if (mo >= POS_OV) lab = 1.0f;
  labels[(size_t)b * N + i] = lab;
}

// ---------------------------------------------------------------------------
// K3: exact JAX RNG — u_f / u_b per (batch, anchor)
//   key(42) --split(32)--> key_b --split(2)--> (kf, kb) --uniform(N)--> u
// ---------------------------------------------------------------------------
__global__ void k_rng(float* __restrict__ u_f, float* __restrict__ u_b, int N) {
  const int b = blockIdx.y;
  const int i = blockIdx.x * blockDim.x + threadIdx.x;
  if (i >= N) return;

  // split(key(42)=(0,42), 32): counts=[0..63]; block j pairs (j, j+32);
  // reshape(32,2) of concat(y0s,y1s) -> b<16: key=(y0[2b],y0[2b+1]),
  //                                    b>=16: key=(y1[2b-32],y1[2b-31])
  unsigned i0 = (b < 16) ? (unsigned)(2 * b) : (unsigned)(2 * b - 32);
  unsigned x0 = i0,      x1 = i0 + 32u;  tf2x32(0u, 42u, x0, x1);
  unsigned y0a = x0, y1a = x1;
  unsigned x2 = i0 + 1u, x3 = i0 + 33u;  tf2x32(0u, 42u, x2, x3);
  unsigned y0b = x2, y1b = x3;
  unsigned e0 = (b < 16) ? y0a : y1a;
  unsigned e1 = (b < 16) ? y0b : y1b;

  // split(key_b, 2): counts=[0,1,2,3]; blocks (0,2),(1,3); kf=(y0,y0), kb=(y1,y1)
  unsigned p0 = 0u, q0 = 2u;  tf2x32(e0, e1, p0, q0);
  unsigned p1 = 1u, q1 = 3u;  tf2x32(e0, e1, p1, q1);
  // kf=(p0,p1), kb=(q0,q1)

  u_f[(size_t)b * N + i] = tf_uniform(p0, p1, i, N);
  u_b[(size_t)b * N + i] = tf_uniform(q0, q1, i, N);
}

// ---------------------------------------------------------------------------
// K4: class histograms of u for top-K selection
// ---------------------------------------------------------------------------
__global__ void k_hist(const float* __restrict__ labels,
                       const float* __restrict__ u_f,
                       const float* __restrict__ u_b,
                       int* __restrict__ hist, int N) {
  const int b = blockIdx.y;
  const int i = blockIdx.x * blockDim.x + threadIdx.x;
  if (i >= N) return;
  float lab = labels[(size_t)b * N + i];
  if (lab == 1.0f) {
    int bin = min((int)(u_f[(size_t)b * N + i] * (float)NBIN), NBIN - 1);
    atomicAdd(&hist[b * 2 * NBIN + bin], 1);
  } else if (lab == 0.0f) {
    int bin = min((int)(u_b[(size_t)b * N + i] * (float)NBIN), NBIN - 1);
    atomicAdd(&hist[b * 2 * NBIN + NBIN + bin], 1);
  }
}

// ---------------------------------------------------------------------------
// K5: per-batch threshold-bin selection.  sel[b] = {t_f, R_f, t_b, R_b};
// t == -1 means "keep all"; within bin t keep the R highest (stable ties).
// ---------------------------------------------------------------------------
__global__ void k_select(const int* __restrict__ hist, int* __restrict__ sel) {
  const int b = blockIdx.x;
  if (threadIdx.x != 0) return;
  const int* hf = hist + b * 2 * NBIN;
  const int* hb = hf + NBIN;

  int total = 0;
  for (int j = 0; j < NBIN; ++j) total += hf[j];
  int t = -1, R = 0;
  if (total > NUM_FG) {
    int acc = 0;
    for (int bin = NBIN - 1; bin >= 0; --bin) {
      if (acc + hf[bin] >= NUM_FG) { t = bin; R = NUM_FG - acc; break; }
      acc += hf[bin];
    }
  }
  sel[b * 4 + 0] = t;
  sel[b * 4 + 1] = R;

  int fg_kept = min(total, NUM_FG);
  int Kb = RPN_BS - fg_kept;
  int totb = 0;
  for (int j = 0; j < NBIN; ++j) totb += hb[j];
  int tb = -1, Rb = 0;
  if (totb > Kb) {
    int acc = 0;
    for (int bin = NBIN - 1; bin >= 0; --bin) {
      if (acc + hb[bin] >= Kb) { tb = bin; Rb = Kb - acc; break; }
      acc += hb[bin];
    }
  }
  sel[b * 4 + 2] = tb;
  sel[b * 4 + 3] = Rb;
}

// ---------------------------------------------------------------------------
// K6: apply subsampling.  Only threshold-bin members need exact stable rank
// (descending u, ascending index ties) — scan is O(N) for those few anchors.
// ---------------------------------------------------------------------------
__global__ void k_subsample(const float* __restrict__ labels,
                            const float* __restrict__ u_f,
                            const float* __restrict__ u_b,
                            const int* __restrict__ sel,
                            float* __restrict__ labels2, int N) {
  const int b = blockIdx.y;
  const int i = blockIdx.x * blockDim.x + threadIdx.x;
  if (i >= N) return;
  float lab = labels[(size_t)b * N + i];
  if (lab == 1.0f || lab == 0.0f) {
    const int cls = (lab == 1.0f) ? 0 : 1;
    const int t = sel[b * 4 + cls * 2 + 0];
    const int R = sel[b * 4 + cls * 2 + 1];
    const float* u = cls ? u_b : u_f;
    const float want = cls ? 0.0f : 1.0f;
    if (t >= 0) {
      float ui = u[(size_t)b * N + i];
      int bin = min((int)(ui * (float)NBIN), NBIN - 1);
      if (bin < t) {
        lab = -1.0f;
      } else if (bin == t) {
        int cnt = 0;
        for (int j = 0; j < N; ++j) {
          if (labels[(size_t)b * N + j] == want) {
            float uj = u[(size_t)b * N + j];
            int bj = min((int)(uj * (float)NBIN), NBIN - 1);
            if (bj == t && (uj > ui || (uj == ui && j < i))) ++cnt;
          }
        }
        if (cnt >= R) lab = -1.0f;
      }
    }
  }
  labels2[(size_t)b * N + i] = lab;
}

// ---------------------------------------------------------------------------
// K7: bbox regression targets + scatter labels/targets to full anchor grid
// ---------------------------------------------------------------------------
__global__ void k_final(const float* __restrict__ gt,
                        const float* __restrict__ anchors,
                        const int* __restrict__ amax,
                        const float* __restrict__ labels2,
                        const int* __restrict__ inds,
                        float* __restrict__ out, int N) {
  const int b = blockIdx.y;
  const int i = blockIdx.x * blockDim.x + threadIdx.x;
  if (i >= N) return;
  const int ind = inds[i];

  out[(size_t)b * TOTAL_ANCH + ind] = labels2[(size_t)b * N + i];

  float a0 = anchors[4 * i + 0], a1 = anchors[4 * i + 1];
  float a2 = anchors[4 * i + 2], a3 = anchors[4 * i + 3];
  float aw = a2 - a0 + 1.0f, ah = a3 - a1 + 1.0f;
  float a_cx = a0 + 0.5f * (aw - 1.0f);
  float a_cy = a1 + 0.5f * (ah - 1.0f);

  const float* g = gt + (size_t)b * GG * 5 + (size_t)amax[(size_t)b * N + i] * 5;
  float g0 = g[0], g1 = g[1], g2 = g[2], g3 = g[3];
  float sgw = g2 - g0 + 1.0f, sgh = g3 - g1 + 1.0f;
  float g_cx = g0 + 0.5f * (sgw - 1.0f);
  float g_cy = g1 + 0.5f * (sgh - 1.0f);

  float* tg = out + (size_t)BB * TOTAL_ANCH + ((size_t)b * TOTAL_ANCH + ind) * 4;
  tg[0] = (g_cx - a_cx) / aw;
  tg[1] = (g_cy - a_cy) / ah;
  tg[2] = logf(sgw / aw);
  tg[3] = logf(sgh / ah);
}

// ---------------------------------------------------------------------------
// launch
// ---------------------------------------------------------------------------
extern "C" void kernel_launch(void* const* d_in, const int* in_sizes, int n_in,
                              void* d_out, int out_size, void* d_ws, size_t ws_size,
                              hipStream_t stream) {
  const float* gt      = (const float*)d_in[0];  // (32,50,5)
  const float* anchors = (const float*)d_in[1];  // (Nin,4)
  const int*   inds    = (const int*)d_in[2];    // (Nin,)
  const int N = in_sizes[2];                     // Nin
  float* out = (float*)d_out;

  // workspace carve-up (all 4-byte typed)
  char* w = (char*)d_ws;
  float*    maxov   = (float*)w;     w += (size_t)BB * N * 4;
  int*      amax    = (int*)w;       w += (size_t)BB * N * 4;
  float*    labels  = (float*)w;     w += (size_t)BB * N * 4;
  float*    labels2 = (float*)w;     w += (size_t)BB * N * 4;
  float*    u_f     = (float*)w;     w += (size_t)BB * N * 4;
  float*    u_b     = (float*)w;     w += (size_t)BB * N * 4;
  unsigned* gtmax   = (unsigned*)w;  w += (size_t)BB * GG * 4;
  int*      hist    = (int*)w;       w += (size_t)BB * 2 * NBIN * 4;
  int*      sel     = (int*)w;

  const dim3 blk(256);
  const dim3 grd((N + 255) / 256, BB);
  const int init_blocks = (out_size + 255) / 256;

  k_init<<<init_blocks, blk, 0, stream>>>(out, out_size, gtmax, hist);
  k_overlap<<<grd, blk, 0, stream>>>(gt, anchors, maxov, amax, gtmax, N);
  k_labels<<<grd, blk, 0, stream>>>(gt, anchors, maxov, gtmax, labels, N);
  k_rng<<<grd, blk, 0, stream>>>(u_f, u_b, N);
  k_hist<<<grd, blk, 0, stream>>>(labels, u_f, u_b, hist, N);
  k_select<<<BB, 64, 0, stream>>>(hist, sel);
  k_subsample<<<grd, blk, 0, stream>>>(labels, u_f, u_b, sel, labels2, N);
  k_final<<<grd, blk, 0, stream>>>(gt, anchors, amax, labels2, inds, out, N);
}